// CRF_4277787427617
// MI455X (gfx1250) — compile-verified
//
#include <hip/hip_runtime.h>
#include <hip/hip_bf16.h>
#include <stdint.h>

// ---------------------------------------------------------------------------
// Viterbi decode (linear-chain CRF), B=128, T=4096, K=64, on gfx1250.
//
// Strategy (MI455X / CDNA5):
//  * One workgroup per batch element (128 blocks, 64 threads = 2 wave32).
//  * Lane j owns tag j. Transition column trans[:, j] lives in 64 VGPRs.
//  * State vector (64 f32) ping-pongs between two LDS buffers; one
//    workgroup barrier per time step.
//  * Emissions streamed HBM -> LDS with double-buffered
//    global_load_async_to_lds_b128 (ASYNCcnt + s_wait_asynccnt).
//  * Backpointers (T-1)*64 u8 = 256KB kept ENTIRELY in LDS (CDNA5: 320KB/WGP)
//    -> backtrack is a dependent ds_load_u8 chain, never touches HBM.
// ---------------------------------------------------------------------------

namespace {
constexpr int B_ = 128;
constexpr int T_ = 4096;
constexpr int K_ = 64;
constexpr int CHUNK = 32;                       // time steps per emission chunk
constexpr int NCHUNK = T_ / CHUNK;              // 128
constexpr int BP_BYTES = (T_ - 1) * K_;         // 262080, 16B aligned
constexpr int SMEM_BYTES = BP_BYTES            // backpointers
                         + 2 * K_ * 4           // state ping-pong
                         + 2 * CHUNK * K_ * 4;  // emission double buffer
} // namespace

#if __has_builtin(__builtin_amdgcn_global_load_async_to_lds_b128)
#define HAVE_ASYNC_LDS 1
#else
#define HAVE_ASYNC_LDS 0
#endif

template <int N>
__device__ __forceinline__ void wait_asynccnt() {
#if HAVE_ASYNC_LDS
#if __has_builtin(__builtin_amdgcn_s_wait_asynccnt)
  __builtin_amdgcn_s_wait_asynccnt(N);
#else
  asm volatile("s_wait_asynccnt %0" ::"i"(N) : "memory");
#endif
#endif
}

// Copy one emission chunk (CHUNK*K_ floats = 512 x 16B) global -> LDS.
// 64 lanes x 8 x b128. Tracked on ASYNCcnt (8 per wave per chunk).
__device__ __forceinline__ void stream_chunk_to_lds(const float* __restrict__ gsrc,
                                                    float* lds_dst, int lane) {
#if HAVE_ASYNC_LDS
  typedef int v4i_t __attribute__((ext_vector_type(4)));
  typedef __attribute__((address_space(1))) v4i_t gv4i_t;
  typedef __attribute__((address_space(3))) v4i_t lv4i_t;
  gv4i_t* g = (gv4i_t*)gsrc;
  lv4i_t* l = (lv4i_t*)lds_dst;
#pragma unroll
  for (int k = 0; k < 8; ++k) {
    const int e = k * 64 + lane;  // 16-byte element index
    __builtin_amdgcn_global_load_async_to_lds_b128(g + e, l + e, 0, 0);
  }
#else
  const float4* g4 = (const float4*)gsrc;
  float4* l4 = (float4*)lds_dst;
#pragma unroll
  for (int k = 0; k < 8; ++k) l4[k * 64 + lane] = g4[k * 64 + lane];
#endif
}

extern "C" __global__ void __launch_bounds__(K_, 1)
viterbi_crf_kernel(const float* __restrict__ pot,     // (B,T,K) f32
                   const float* __restrict__ trans,   // (K,K)   f32
                   const int* __restrict__ mask,      // (B,T)   i32
                   int* __restrict__ out)             // B*T tags ++ B lengths
{
  extern __shared__ char smem[];
  uint8_t* bpm    = (uint8_t*)smem;                         // (T-1) rows x 64
  float*   stateA = (float*)(smem + BP_BYTES);
  float*   stateB = stateA + K_;
  float*   emit0  = stateB + K_;
  float*   emit1  = emit0 + CHUNK * K_;

  const int b = blockIdx.x;
  const int j = threadIdx.x;  // tag owned by this lane, 0..63
  const float* potB = pot + (size_t)b * T_ * K_;

  // Transition column trans[:, j] -> 64 VGPRs (coalesced: lanes j contiguous).
  float tc[K_];
#pragma unroll
  for (int i = 0; i < K_; ++i) tc[i] = trans[i * K_ + j];

  // state_0 = potentials[b, 0, :]
  stateA[j] = potB[j];

  // Prime the async pipeline: chunks 0 and 1 in flight.
  stream_chunk_to_lds(potB + 0 * CHUNK * K_, emit0, j);
  stream_chunk_to_lds(potB + 1 * CHUNK * K_, emit1, j);

  float* cur = stateA;
  float* nxt = stateB;

#pragma unroll 1
  for (int c = 0; c < NCHUNK; ++c) {
    float* ebuf = (c & 1) ? emit1 : emit0;
    // Oldest 8 async ops (= this chunk) must be complete; up to 8 newer
    // (next chunk) may stay in flight.
    if (c + 1 < NCHUNK) wait_asynccnt<8>(); else wait_asynccnt<0>();
    __syncthreads();  // cross-wave visibility of LDS (emissions + state)

    const int t0 = c * CHUNK;
    const int ts = (c == 0) ? 1 : 0;  // t=0 is the init row, no transition

#pragma unroll 1
    for (int tt = ts; tt < CHUNK; ++tt) {
      const int t = t0 + tt;
      const float4* s4 = (const float4*)cur;
      // 4 independent (max, argmax) chains for ILP.
      float m0 = -3.0e38f, m1 = -3.0e38f, m2 = -3.0e38f, m3 = -3.0e38f;
      int i0 = 0, i1 = 1, i2 = 2, i3 = 3;
#pragma unroll
      for (int q = 0; q < 16; ++q) {
        const float4 s = s4[q];
        const float v0 = s.x + tc[4 * q + 0];
        const float v1 = s.y + tc[4 * q + 1];
        const float v2 = s.z + tc[4 * q + 2];
        const float v3 = s.w + tc[4 * q + 3];
        if (v0 > m0) { m0 = v0; i0 = 4 * q + 0; }
        if (v1 > m1) { m1 = v1; i1 = 4 * q + 1; }
        if (v2 > m2) { m2 = v2; i2 = 4 * q + 2; }
        if (v3 > m3) { m3 = v3; i3 = 4 * q + 3; }
      }
      float m = m0; int bi = i0;
      if (m1 > m) { m = m1; bi = i1; }
      if (m2 > m) { m = m2; bi = i2; }
      if (m3 > m) { m = m3; bi = i3; }

      nxt[j] = m + ebuf[tt * K_ + j];
      bpm[(t - 1) * K_ + j] = (uint8_t)bi;

      float* tmp = cur; cur = nxt; nxt = tmp;
      __syncthreads();
    }
    // Emission buffer for chunk c is now dead for every wave (they all passed
    // the last per-step barrier) -> refill it with chunk c+2.
    if (c + 2 < NCHUNK) stream_chunk_to_lds(potB + (size_t)(c + 2) * CHUNK * K_, ebuf, j);
  }

  // ---- sequence_lengths[b] = sum(mask[b, :]) (all 64 lanes cooperate) ----
  {
    const int* mrow = mask + (size_t)b * T_;
    int partial = 0;
#pragma unroll 4
    for (int t = j; t < T_; t += K_) partial += mrow[t];
    ((int*)nxt)[j] = partial;  // nxt = stale state buffer, safe scratch
    __syncthreads();
    if (j == 0) {
      int s = 0;
#pragma unroll
      for (int i = 0; i < K_; ++i) s += ((int*)nxt)[i];
      out[B_ * T_ + b] = s;
    }
  }

  // ---- backtrack entirely from LDS-resident backpointers ----
  if (j == 0) {
    float m = cur[0];
    int tag = 0;
#pragma unroll
    for (int i = 1; i < K_; ++i) {
      const float v = cur[i];
      if (v > m) { m = v; tag = i; }
    }
    int* trow = out + (size_t)b * T_;
    trow[T_ - 1] = tag;
#pragma unroll 1
    for (int t = T_ - 2; t >= 0; --t) {
      tag = (int)bpm[t * K_ + tag];   // dependent ds_load_u8 chain
      trow[t] = tag;
    }
  }
}

extern "C" void kernel_launch(void* const* d_in, const int* in_sizes, int n_in,
                              void* d_out, int out_size, void* d_ws, size_t ws_size,
                              hipStream_t stream) {
  (void)in_sizes; (void)n_in; (void)d_ws; (void)ws_size; (void)out_size;
  const float* pot   = (const float*)d_in[0];   // inputs      (B,T,K) f32
  const float* trans = (const float*)d_in[1];   // transitions (K,K)   f32
  const int*   mask  = (const int*)d_in[2];     // mask        (B,T)   i32
  int* out = (int*)d_out;                        // B*T tags ++ B lengths (i32)

  viterbi_crf_kernel<<<dim3(B_), dim3(K_), SMEM_BYTES, stream>>>(pot, trans, mask, out);
}